// AF_73486890435192
// MI455X (gfx1250) — compile-verified
//
#include <hip/hip_runtime.h>

typedef float v2f __attribute__((ext_vector_type(2)));
typedef float v8f __attribute__((ext_vector_type(8)));

#define BATCH 2
#define CIN   128
#define COUT  128
#define HH_   96
#define WW_   96
#define HW    (HH_ * WW_)
#define KS    7
#define PADR  3

// ---------------------------------------------------------------------------
// Kernel 1: dual GEMM  Q = Wq * X,  K = Wk * X  using V_WMMA_F32_16X16X4_F32.
// X is [Cin, B*HW] (row stride HW within a batch), W is [Cout, Cin] row-major.
// One wave computes a 16(cout) x 16(pixel) tile of BOTH Q and K, sharing the
// B-matrix (x) fragment. 8 waves/block cover all 128 couts for one 16-pixel
// strip. K=128 -> 32 wmma steps of K=4, exact fp32 (no precision loss).
// ---------------------------------------------------------------------------
__global__ __launch_bounds__(256) void AF_gemm_qk(
    const float* __restrict__ x,
    const float* __restrict__ wq,
    const float* __restrict__ wk,
    float* __restrict__ qout,
    float* __restrict__ kout) {
  const int lane = threadIdx.x & 31;
  const int wave = threadIdx.x >> 5;
  const int n0   = blockIdx.x * 16;          // flat spatial index over B*HW
  const int b    = n0 / HW;
  const int nb   = n0 - b * HW;
  const int cobase = wave * 16;

  const float* xb    = x + (size_t)b * CIN * HW + nb;
  const int    nlane = lane & 15;
  const int    koff  = (lane < 16) ? 0 : 2;  // half-wave K offset (ISA 7.12.2)

  const float* wqrow = wq + (size_t)(cobase + nlane) * CIN;
  const float* wkrow = wk + (size_t)(cobase + nlane) * CIN;

  v8f accq = {};
  v8f acck = {};

  for (int kb = 0; kb < CIN; kb += 4) {
    const int kk = kb + koff;
    v2f aq, ak, bv;
    // A fragments: lane m = co row, v0 = K(kk), v1 = K(kk+1)
    aq.x = wqrow[kk];
    aq.y = wqrow[kk + 1];
    ak.x = wkrow[kk];
    ak.y = wkrow[kk + 1];
    // B fragment: lane n = pixel col, v0 = row kk, v1 = row kk+1 of X
    bv.x = xb[(size_t)kk * HW + nlane];
    bv.y = xb[(size_t)(kk + 1) * HW + nlane];

    accq = __builtin_amdgcn_wmma_f32_16x16x4_f32(
        /*neg_a=*/false, aq, /*neg_b=*/false, bv,
        /*c_mod=*/(short)0, accq, /*reuse_a=*/false, /*reuse_b=*/false);
    acck = __builtin_amdgcn_wmma_f32_16x16x4_f32(
        /*neg_a=*/false, ak, /*neg_b=*/false, bv,
        /*c_mod=*/(short)0, acck, /*reuse_a=*/false, /*reuse_b=*/false);
  }

  // C/D layout: VGPR r -> M = r (lanes 0-15) / M = 8+r (lanes 16-31), N = lane&15
  const int mbase = (lane < 16) ? 0 : 8;
#pragma unroll
  for (int r = 0; r < 8; ++r) {
    const int co = cobase + mbase + r;
    const size_t off = ((size_t)b * COUT + co) * HW + nb + nlane;
    qout[off] = accq[r];
    kout[off] = acck[r];
  }
}

// ---------------------------------------------------------------------------
// Kernel 2: per-channel 7x7 windowed softmax attention.
// Block = one (b, c, h) row; 96 threads (3 wave32) over w.
// Stage 7 x 102 halo rows of k and v in LDS (zero for out-of-bounds, matching
// the reference's zero-pad: OOB taps contribute exp(q*rel) to the denominator
// with v=0). rel_h biases channels [0,64) over kh; rel_w biases [64,128) over kw.
// ---------------------------------------------------------------------------
__global__ __launch_bounds__(96) void AF_attn(
    const float* __restrict__ q,
    const float* __restrict__ kf,
    const float* __restrict__ v,
    const float* __restrict__ rel_h,
    const float* __restrict__ rel_w,
    float* __restrict__ out) {
  const int w = threadIdx.x;              // 0..95
  int bch = blockIdx.x;
  const int h = bch % HH_; bch /= HH_;
  const int c = bch % COUT;
  const int b = bch / COUT;

  __shared__ float sk[KS][104];           // 102 used, stride 104
  __shared__ float sv[KS][104];

  const size_t base = ((size_t)b * COUT + c) * HW;

  // cooperative halo load: rows h-3..h+3, cols -3..98
  for (int i = threadIdx.x; i < KS * 102; i += 96) {
    const int r  = i / 102;
    const int cc = i - r * 102;
    const int hh = h - PADR + r;
    const int ww = cc - PADR;
    float kvv = 0.0f, vvv = 0.0f;
    if (hh >= 0 && hh < HH_ && ww >= 0 && ww < WW_) {
      const size_t g = base + (size_t)hh * WW_ + ww;
      kvv = kf[g];
      vvv = v[g];
    }
    sk[r][cc] = kvv;
    sv[r][cc] = vvv;
  }
  __syncthreads();

  const float qv = q[base + (size_t)h * WW_ + w];

  const bool use_h = (c < COUT / 2);
  float rel[KS];
#pragma unroll
  for (int i = 0; i < KS; ++i)
    rel[i] = use_h ? rel_h[c * KS + i] : rel_w[(c - COUT / 2) * KS + i];

  // pass 1: logits + max (fully unrolled -> registers)
  float lg[KS * KS];
  float m = -3.4e38f;
#pragma unroll
  for (int kh = 0; kh < KS; ++kh) {
#pragma unroll
    for (int kw = 0; kw < KS; ++kw) {
      const float rb = use_h ? rel[kh] : rel[kw];
      const float l  = qv * (sk[kh][w + kw] + rb);
      lg[kh * KS + kw] = l;
      m = fmaxf(m, l);
    }
  }

  // pass 2: exp-sum and weighted v accumulation
  float s = 0.0f, acc = 0.0f;
#pragma unroll
  for (int kh = 0; kh < KS; ++kh) {
#pragma unroll
    for (int kw = 0; kw < KS; ++kw) {
      const float e = __expf(lg[kh * KS + kw] - m);
      s   += e;
      acc += e * sv[kh][w + kw];
    }
  }

  out[base + (size_t)h * WW_ + w] = acc / s;
}

// ---------------------------------------------------------------------------
extern "C" void kernel_launch(void* const* d_in, const int* in_sizes, int n_in,
                              void* d_out, int out_size, void* d_ws, size_t ws_size,
                              hipStream_t stream) {
  const float* x     = (const float*)d_in[0];   // [B, Cin, H, W]
  const float* v     = (const float*)d_in[1];   // [B, Cout, H, W]
  const float* wq    = (const float*)d_in[2];   // [Cout, Cin]
  const float* wk    = (const float*)d_in[3];   // [Cout, Cin]
  const float* rel_h = (const float*)d_in[4];   // [Cout/2, 7]
  const float* rel_w = (const float*)d_in[5];   // [Cout/2, 7]
  float* out = (float*)d_out;

  float* qbuf = (float*)d_ws;                            // B*Cout*HW floats
  float* kbuf = qbuf + (size_t)BATCH * COUT * HW;        // B*Cout*HW floats

  // Stage 1: WMMA dual GEMM -> q, k in workspace
  const int n_strips = (BATCH * HW) / 16;                // 1152 blocks
  AF_gemm_qk<<<n_strips, 256, 0, stream>>>(x, wq, wk, qbuf, kbuf);

  // Stage 2: fused windowed softmax-attention
  AF_attn<<<BATCH * COUT * HH_, 96, 0, stream>>>(qbuf, kbuf, v, rel_h, rel_w, out);
}